// Advmmd_73289321939443
// MI455X (gfx1250) — compile-verified
//
#include <hip/hip_runtime.h>
#include <math.h>

typedef __attribute__((ext_vector_type(2))) float v2f;
typedef __attribute__((ext_vector_type(8))) float v8f;

#define B_ROWS 4096
#define N_TOT  8192
#define D_DIM  2048
#define H_DIM  128

#define BT 64          // block tile (M and N)
#define BK 32          // K chunk staged in LDS
#define LDS_STRIDE 36  // 36 ≡ 0 mod 4 -> 16B-aligned b128 stores; conflict-free b64 frag loads
#define TILE_F (BT * LDS_STRIDE)

// workspace layout (floats)
#define WS_ADV 0
#define WS_MMD 1
#define WS_C   2
#define WS_NIB 3
#define WS_V   4
#define WS_CS  (WS_V + D_DIM)
#define WS_SQ  (WS_CS + D_DIM)

__device__ __forceinline__ const float* row_ptr(const float* src, const float* tgt, int row) {
    return (row < B_ROWS) ? (src + (size_t)row * D_DIM)
                          : (tgt + (size_t)(row - B_ROWS) * D_DIM);
}

// ---- zero the accumulators (graph replays re-use ws) ----
__global__ void k_init(float* ws) {
    if (threadIdx.x < 2) ws[threadIdx.x] = 0.f;
}

// ---- fold discriminator: v[d] = sum_k W2[k]*W1[k][d]; c = b1.W2 + b2 ----
__global__ void k_fold(const float* __restrict__ W1, const float* __restrict__ b1,
                       const float* __restrict__ W2, const float* __restrict__ b2,
                       float* __restrict__ ws) {
    int d = blockIdx.x * blockDim.x + threadIdx.x;
    if (d < D_DIM) {
        float acc = 0.f;
        for (int k = 0; k < H_DIM; ++k) acc += W2[k] * W1[k * D_DIM + d];
        ws[WS_V + d] = acc;
    }
    if (blockIdx.x == 0 && threadIdx.x == 0) {
        float c = b2[0];
        for (int k = 0; k < H_DIM; ++k) c += b1[k] * W2[k];
        ws[WS_C] = c;
    }
}

// ---- column sums of T = concat(source,target): needed for closed-form sum(L2) ----
__global__ void k_colsum(const float* __restrict__ src, const float* __restrict__ tgt,
                         float* __restrict__ ws) {
    int d = blockIdx.x * blockDim.x + threadIdx.x;
    if (d >= D_DIM) return;
    float acc = 0.f;
    for (int i = 0; i < B_ROWS; ++i) acc += src[(size_t)i * D_DIM + d];
    for (int i = 0; i < B_ROWS; ++i) acc += tgt[(size_t)i * D_DIM + d];
    ws[WS_CS + d] = acc;
}

// ---- per-row: sq[i] = ||x_i||^2; logit = x_i.v + c; accumulate adv loss ----
__global__ void k_rowstats(const float* __restrict__ src, const float* __restrict__ tgt,
                           float* __restrict__ ws) {
    const int wave = threadIdx.x >> 5;
    const int lane = threadIdx.x & 31;
    const int row  = blockIdx.x * 8 + wave;
    if (row >= N_TOT) return;
    const float* p   = row_ptr(src, tgt, row);
    const float4* p4 = (const float4*)p;
    const float4* v4 = (const float4*)(ws + WS_V);
    float dot = 0.f, sq = 0.f;
    for (int c = lane; c < D_DIM / 4; c += 32) {
        float4 x = p4[c], v = v4[c];
        dot += x.x * v.x + x.y * v.y + x.z * v.z + x.w * v.w;
        sq  += x.x * x.x + x.y * x.y + x.z * x.z + x.w * x.w;
    }
    for (int m = 16; m >= 1; m >>= 1) {
        dot += __shfl_xor(dot, m, 32);
        sq  += __shfl_xor(sq,  m, 32);
    }
    if (lane == 0) {
        ws[WS_SQ + row] = sq;
        float z  = dot + ws[WS_C];
        float zz = (row < B_ROWS) ? z : -z;           // src: ls(z), tgt: ls(-z)
        float ls = fminf(zz, 0.f) - __logf(1.f + __expf(-fabsf(zz)));
        atomicAdd(&ws[WS_ADV], -ls);
    }
}

// ---- bandwidth from closed form: sum(L2) = 2n*sum(sq) - 2*||colsum||^2 ----
__global__ void k_prep(float* __restrict__ ws) {
    __shared__ float red[256];
    float a = 0.f;
    for (int d = threadIdx.x; d < D_DIM; d += 256) { float c = ws[WS_CS + d]; a += c * c; }
    float b = 0.f;
    for (int i = threadIdx.x; i < N_TOT; i += 256) b += ws[WS_SQ + i];
    red[threadIdx.x] = a; __syncthreads();
    for (int s = 128; s > 0; s >>= 1) { if (threadIdx.x < s) red[threadIdx.x] += red[threadIdx.x + s]; __syncthreads(); }
    float g2 = red[0]; __syncthreads();
    red[threadIdx.x] = b; __syncthreads();
    for (int s = 128; s > 0; s >>= 1) { if (threadIdx.x < s) red[threadIdx.x] += red[threadIdx.x + s]; __syncthreads(); }
    if (threadIdx.x == 0) {
        double sumsq = (double)red[0];
        double bwsum = 2.0 * (double)N_TOT * sumsq - 2.0 * (double)g2;
        double bandwidth = bwsum / ((double)N_TOT * (double)N_TOT - (double)N_TOT);
        bandwidth *= 0.25;                 // / KERNEL_MUL^(KERNEL_NUM//2) = /4
        // largest denominator: bandwidth * 2^4; terms are a, a^2, a^4, a^8, a^16
        ws[WS_NIB] = (float)(-1.0 / (bandwidth * 16.0));
    }
}

// ---- async global->LDS tile stage (no VGPR data; tracked by ASYNCcnt) ----
// 8 async b128 instructions per thread per K-chunk (4 for A tile, 4 for B tile).
__device__ __forceinline__ void issue_tile_loads(const float* src, const float* tgt,
                                                 float* ldsA, float* ldsB,
                                                 int iBase, int jBase, int kb, int tid) {
#pragma unroll
    for (int rr = 0; rr < 4; ++rr) {
        int idx = tid + 128 * rr;      // 0..511 b128 slots
        int row = idx >> 3;            // 0..63
        int c4  = (idx & 7) << 2;      // 0,4,...,28
        const float* pA = row_ptr(src, tgt, iBase + row) + kb + c4;
        const float* pB = row_ptr(src, tgt, jBase + row) + kb + c4;
        // generic LDS pointer: low 32 bits are the LDS byte offset (ISA aperture rule)
        unsigned la = (unsigned)(size_t)(ldsA + row * LDS_STRIDE + c4);
        unsigned lb = (unsigned)(size_t)(ldsB + row * LDS_STRIDE + c4);
        asm volatile("global_load_async_to_lds_b128 %0, %1, off"
                     :: "v"(la), "v"(pA) : "memory");
        asm volatile("global_load_async_to_lds_b128 %0, %1, off"
                     :: "v"(lb), "v"(pB) : "memory");
    }
}

// ---- main WMMA kernel: lower-triangular 64x64 tiles of G = T T^T, fused MMD epilogue ----
// Double-buffered LDS; async loads overlap the WMMA phase of the previous K-chunk.
__global__ __launch_bounds__(128) void k_mmd(const float* __restrict__ src,
                                             const float* __restrict__ tgt,
                                             float* __restrict__ ws) {
    __shared__ float ldsA[2][TILE_F];
    __shared__ float ldsB[2][TILE_F];
    __shared__ float red[128];

    // decode linear block id -> (bi, bj), bi >= bj
    int L  = (int)blockIdx.x;
    int bi = (int)((sqrtf(8.f * (float)L + 1.f) - 1.f) * 0.5f);
    while ((bi + 1) * (bi + 2) / 2 <= L) ++bi;
    while (bi * (bi + 1) / 2 > L) --bi;
    int bj = L - bi * (bi + 1) / 2;
    const int iBase = bi * BT, jBase = bj * BT;

    const int tid    = threadIdx.x;
    const int waveId = tid >> 5;
    const int lane   = tid & 31;
    const int half   = lane >> 4;   // K pair / row-half select
    const int r16    = lane & 15;
    const int wm     = (waveId & 1) * 32;   // wave M offset in block tile
    const int wn     = (waveId >> 1) * 32;  // wave N offset

    v8f acc[2][2] = {};

    // prologue: stage chunk 0 into buffer 0
    issue_tile_loads(src, tgt, ldsA[0], ldsB[0], iBase, jBase, 0, tid);

    int cur = 0;
    for (int kb = 0; kb < D_DIM; kb += BK) {
        const bool has_next = (kb + BK) < D_DIM;
        if (has_next) {
            // prefetch next chunk into the spare buffer (8 async instrs)
            issue_tile_loads(src, tgt, ldsA[cur ^ 1], ldsB[cur ^ 1], iBase, jBase, kb + BK, tid);
            // <=8 outstanding ==> the current chunk's 8 loads have completed (in-order)
            asm volatile("s_wait_asynccnt 0x8" ::: "memory");
        } else {
            asm volatile("s_wait_asynccnt 0x0" ::: "memory");
        }
        __syncthreads();   // current buffer now valid for all waves

        const float* lA = ldsA[cur];
        const float* lB = ldsB[cur];
#pragma unroll
        for (int kk = 0; kk < BK; kk += 4) {
            int k2 = kk + half * 2;    // lanes 0-15 -> K=k,k+1 ; lanes 16-31 -> K=k+2,k+3
            v2f afr[2], bfr[2];
#pragma unroll
            for (int t = 0; t < 2; ++t) {
                afr[t] = *(const v2f*)&lA[(wm + t * 16 + r16) * LDS_STRIDE + k2];
                bfr[t] = *(const v2f*)&lB[(wn + t * 16 + r16) * LDS_STRIDE + k2];
            }
#pragma unroll
            for (int tm = 0; tm < 2; ++tm)
#pragma unroll
                for (int tn = 0; tn < 2; ++tn)
                    acc[tm][tn] = __builtin_amdgcn_wmma_f32_16x16x4_f32(
                        false, afr[tm], false, bfr[tn], (short)0, acc[tm][tn], false, false);
        }
        __syncthreads();   // all waves done reading; safe to overwrite this buffer next round
        cur ^= 1;
    }

    // fused epilogue: L2 -> clamp -> 5-kernel sum via repeated squaring -> signed accumulate
    const float nib = ws[WS_NIB];
    const float* sq = ws + WS_SQ;
    float local = 0.f;
#pragma unroll
    for (int tm = 0; tm < 2; ++tm) {
#pragma unroll
        for (int tn = 0; tn < 2; ++tn) {
            int   j  = jBase + wn + tn * 16 + r16;        // N = lane&15
            float sj = sq[j];
#pragma unroll
            for (int v = 0; v < 8; ++v) {
                int   i  = iBase + wm + tm * 16 + half * 8 + v;  // M = 8*(lane>=16)+vgpr
                float g  = acc[tm][tn][v];
                float l2 = fmaxf(sq[i] + sj - 2.f * g, 0.f);
                float a  = __expf(l2 * nib);
                float s  = a;
                a *= a; s += a;   // a^2
                a *= a; s += a;   // a^4
                a *= a; s += a;   // a^8
                a *= a; s += a;   // a^16
                float sgn = ((i < B_ROWS) == (j < B_ROWS)) ? 1.f : -1.f;
                local += sgn * s;
            }
        }
    }
    if (bi != bj) local *= 2.f;    // symmetric off-diagonal blocks counted twice

    red[tid] = local;
    __syncthreads();
    for (int s = 64; s > 0; s >>= 1) { if (tid < s) red[tid] += red[tid + s]; __syncthreads(); }
    if (tid == 0) atomicAdd(&ws[WS_MMD], red[0]);
}

// ---- finalize ----
__global__ void k_fin(const float* __restrict__ ws, float* __restrict__ out) {
    if (threadIdx.x == 0) {
        out[0] = ws[WS_MMD] / ((float)B_ROWS * (float)B_ROWS);
        out[1] = ws[WS_ADV] / (2.f * (float)B_ROWS);
    }
}

extern "C" void kernel_launch(void* const* d_in, const int* in_sizes, int n_in,
                              void* d_out, int out_size, void* d_ws, size_t ws_size,
                              hipStream_t stream) {
    const float* src = (const float*)d_in[0];
    const float* tgt = (const float*)d_in[1];
    const float* W1  = (const float*)d_in[2];
    const float* b1  = (const float*)d_in[3];
    const float* W2  = (const float*)d_in[4];
    const float* b2  = (const float*)d_in[5];
    float* ws  = (float*)d_ws;
    float* out = (float*)d_out;

    k_init<<<1, 64, 0, stream>>>(ws);
    k_fold<<<D_DIM / 256, 256, 0, stream>>>(W1, b1, W2, b2, ws);
    k_colsum<<<D_DIM / 256, 256, 0, stream>>>(src, tgt, ws);
    k_rowstats<<<N_TOT / 8, 256, 0, stream>>>(src, tgt, ws);
    k_prep<<<1, 256, 0, stream>>>(ws);
    const int NB = N_TOT / BT;
    const int nblk = NB * (NB + 1) / 2;
    k_mmd<<<nblk, 128, 0, stream>>>(src, tgt, ws);
    k_fin<<<1, 64, 0, stream>>>(ws, out);
}